// BERT_CRF_29205777613207
// MI455X (gfx1250) — compile-verified
//
#include <hip/hip_runtime.h>
#include <hip/hip_bf16.h>
#include <math.h>

// Problem constants (B, S, H, T) = (16, 4096, 768, 3)
#define Bsz 16
#define Ssz 4096
#define Hsz 768
#define Tsz 3

#define ROWS   (Bsz * Ssz)        // 65536 rows of the emissions GEMM
#define EMIS_STRIDE 4             // pad T=3 -> 4 for aligned/cheap addressing
#define LDSW_STRIDE 776           // 768 + 8 pad: avoids 64-bank LDS conflicts

#define CHL 64                    // chunk length for parallel log-semiring scan
#define NCH 64                    // ceil((S-1)/CHL) = ceil(4095/64) = 64
#define NEG_INF (-1e30f)

typedef float v2f __attribute__((ext_vector_type(2)));
typedef float v8f __attribute__((ext_vector_type(8)));

// ---------------------------------------------------------------------------
// Kernel 1: emissions[b,s,t] = x[b,s,:] . W[t,:] + bias[t]
// One wave per 16x16 output tile (only 3 columns valid, T=3). Memory bound:
// streams 201 MB of fp32 x once -> ~8.6us floor at 23.3 TB/s.
// Uses V_WMMA_F32_16X16X4_F32 (wave32). W (9KB) staged in LDS, reused by all
// 8 waves x 192 k-steps of the block.
// ---------------------------------------------------------------------------
__global__ __launch_bounds__(256) void crf_emis_wmma(
    const float* __restrict__ x, const float* __restrict__ W,
    const float* __restrict__ bias, float* __restrict__ emis) {
  __shared__ float ldsW[Tsz * LDSW_STRIDE];
  for (int i = threadIdx.x; i < Tsz * Hsz; i += 256) {
    int t = i / Hsz;
    int h = i - t * Hsz;
    ldsW[t * LDSW_STRIDE + h] = W[i];
  }
  __syncthreads();

  const int wave = threadIdx.x >> 5;
  const int lane = threadIdx.x & 31;
  const int tile = blockIdx.x * 8 + wave;     // 4096 tiles total
  const int row0 = tile * 16;
  const int half = lane >> 4;                 // 0: K={0,1}, 1: K={2,3}
  const int m    = lane & 15;                 // A: row within tile; B: column N

  const float* arow = x + (size_t)(row0 + m) * Hsz;
  // All lanes perform the LDS load (EXEC must stay all-1s around WMMA);
  // invalid columns (N >= 3) are zeroed via select, not divergence.
  const float* brow = ldsW + (m < Tsz ? m : 0) * LDSW_STRIDE;
  const bool bvalid = (m < Tsz);

  v8f c = {};  // fp32 accumulator tile (8 VGPRs / lane)

#if __has_builtin(__builtin_amdgcn_wmma_f32_16x16x4_f32)
  #pragma unroll 4
  for (int k = 0; k < Hsz; k += 4) {
    const int kk = k + 2 * half;
    if ((k & 63) == 0)  // stream-ahead prefetch of the A row (global_prefetch_b8)
      __builtin_prefetch(arow + kk + 64, 0, 1);
    const float2 av = *reinterpret_cast<const float2*>(arow + kk);
    const float2 bv = *reinterpret_cast<const float2*>(brow + kk);
    v2f a; a.x = av.x; a.y = av.y;
    v2f b; b.x = bvalid ? bv.x : 0.0f; b.y = bvalid ? bv.y : 0.0f;
    // (neg_a, A, neg_b, B, c_mod, C, reuse_a, reuse_b)
    c = __builtin_amdgcn_wmma_f32_16x16x4_f32(false, a, false, b,
                                              (short)0, c, false, false);
  }
#else
  // Fallback (keeps the round compiling if the f32 WMMA builtin is missing;
  // histogram would show wmma=0 and we switch to the bf16 WMMA path).
  for (int k = 0; k < Hsz; k += 4) {
    const int kk = k + 2 * half;
    const float2 av = *reinterpret_cast<const float2*>(arow + kk);
    const float2 bv = *reinterpret_cast<const float2*>(brow + kk);
    float bx = bvalid ? bv.x : 0.0f, by = bvalid ? bv.y : 0.0f;
    c[0] += av.x * bx + av.y * by;  // placeholder only
  }
#endif

  // D layout: VGPR v, lanes 0-15 -> (M=v, N=lane); lanes 16-31 -> (M=8+v, N=lane-16)
  if (m < Tsz) {
    const float bt = bias[m];
    const int rbase = row0 + half * 8;
    #pragma unroll
    for (int v = 0; v < 8; ++v)
      emis[(size_t)(rbase + v) * EMIS_STRIDE + m] = c[v] + bt;
  }
}

// ---------------------------------------------------------------------------
// log-sum-exp of 3 values
// ---------------------------------------------------------------------------
__device__ __forceinline__ float lse3(float a, float b, float c) {
  float mx = fmaxf(a, fmaxf(b, c));
  return mx + __logf(__expf(a - mx) + __expf(b - mx) + __expf(c - mx));
}

// ---------------------------------------------------------------------------
// Kernel 2: parallel chunking of the CRF forward scan.
// alpha_s = alpha_{s-1} (x) A_s in the log semiring, A_s[i][j] = trans[i][j] +
// emit_s[j] (identity when mask==0). Each thread computes the product of CHL
// consecutive A_s matrices -> serial critical path drops 4095 -> 64.
// ---------------------------------------------------------------------------
__global__ __launch_bounds__(256) void crf_chunk_scan(
    const float* __restrict__ emis, const int* __restrict__ mask,
    const float* __restrict__ trans, float* __restrict__ chunkM) {
  const int tid = blockIdx.x * 256 + threadIdx.x;
  if (tid >= Bsz * NCH) return;
  const int b = tid / NCH;
  const int c = tid - b * NCH;

  float tr[9];
  #pragma unroll
  for (int i = 0; i < 9; ++i) tr[i] = trans[i];

  float M[9];
  #pragma unroll
  for (int i = 0; i < 3; ++i)
    #pragma unroll
    for (int j = 0; j < 3; ++j) M[i * 3 + j] = (i == j) ? 0.0f : NEG_INF;

  const int sbeg = 1 + c * CHL;
  int send = sbeg + CHL; if (send > Ssz) send = Ssz;
  for (int s = sbeg; s < send; ++s) {
    if (mask[b * Ssz + s] != 0) {
      const float* e = emis + (size_t)(b * Ssz + s) * EMIS_STRIDE;
      const float e0 = e[0], e1 = e[1], e2 = e[2];
      float N[9];
      #pragma unroll
      for (int i = 0; i < 3; ++i) {
        const float mi0 = M[i*3+0], mi1 = M[i*3+1], mi2 = M[i*3+2];
        N[i*3+0] = lse3(mi0 + tr[0], mi1 + tr[3], mi2 + tr[6]) + e0;
        N[i*3+1] = lse3(mi0 + tr[1], mi1 + tr[4], mi2 + tr[7]) + e1;
        N[i*3+2] = lse3(mi0 + tr[2], mi1 + tr[5], mi2 + tr[8]) + e2;
      }
      #pragma unroll
      for (int i = 0; i < 9; ++i) M[i] = N[i];
    }
  }
  float* out = chunkM + (size_t)tid * 9;
  #pragma unroll
  for (int i = 0; i < 9; ++i) out[i] = M[i];
}

// ---------------------------------------------------------------------------
// Kernel 3: numerator (parallel gather-reduce over S per batch, one wave32
// per batch), chunk-matrix fold for the denominator, final NLL scalar.
// ---------------------------------------------------------------------------
__global__ __launch_bounds__(512) void crf_finalize(
    const float* __restrict__ emis, const int* __restrict__ y,
    const int* __restrict__ mask, const float* __restrict__ start_t,
    const float* __restrict__ end_t, const float* __restrict__ trans,
    const float* __restrict__ chunkM, float* __restrict__ out) {
  __shared__ float llh_sh[Bsz];
  const int wave = threadIdx.x >> 5;   // 16 waves == 16 batches
  const int lane = threadIdx.x & 31;
  const int b = wave;

  float num = 0.0f;
  int msum = 0;
  for (int s = lane; s < Ssz; s += 32) {
    const int mt = mask[b * Ssz + s];
    msum += mt;
    if (s >= 1 && mt != 0) {
      const int yp = y[b * Ssz + s - 1];
      const int yc = y[b * Ssz + s];
      num += trans[yp * 3 + yc] + emis[(size_t)(b * Ssz + s) * EMIS_STRIDE + yc];
    }
  }
  #pragma unroll
  for (int off = 16; off >= 1; off >>= 1) {   // wave32 reduction
    num  += __shfl_down(num, off, 32);
    msum += __shfl_down(msum, off, 32);
  }

  if (lane == 0) {
    const int y0 = y[b * Ssz];
    num += start_t[y0] + emis[(size_t)(b * Ssz) * EMIS_STRIDE + y0];
    const int seq_end = msum - 1;
    num += end_t[y[b * Ssz + seq_end]];

    float al[3];
    #pragma unroll
    for (int j = 0; j < 3; ++j)
      al[j] = start_t[j] + emis[(size_t)(b * Ssz) * EMIS_STRIDE + j];
    for (int c = 0; c < NCH; ++c) {
      const float* M = chunkM + (size_t)(b * NCH + c) * 9;
      const float n0 = lse3(al[0] + M[0], al[1] + M[3], al[2] + M[6]);
      const float n1 = lse3(al[0] + M[1], al[1] + M[4], al[2] + M[7]);
      const float n2 = lse3(al[0] + M[2], al[1] + M[5], al[2] + M[8]);
      al[0] = n0; al[1] = n1; al[2] = n2;
    }
    const float denom = lse3(al[0] + end_t[0], al[1] + end_t[1], al[2] + end_t[2]);
    llh_sh[b] = num - denom;
  }
  __syncthreads();
  if (threadIdx.x == 0) {
    float s = 0.0f;
    #pragma unroll
    for (int i = 0; i < Bsz; ++i) s += llh_sh[i];
    out[0] = -s / (float)Bsz;
  }
}

// ---------------------------------------------------------------------------
extern "C" void kernel_launch(void* const* d_in, const int* in_sizes, int n_in,
                              void* d_out, int out_size, void* d_ws, size_t ws_size,
                              hipStream_t stream) {
  (void)in_sizes; (void)n_in; (void)out_size; (void)ws_size;
  const float* x       = (const float*)d_in[0];
  const int*   y       = (const int*)  d_in[1];
  const int*   mask    = (const int*)  d_in[2];
  const float* W       = (const float*)d_in[3];
  const float* bias    = (const float*)d_in[4];
  const float* start_t = (const float*)d_in[5];
  const float* end_t   = (const float*)d_in[6];
  const float* trans   = (const float*)d_in[7];
  float* out = (float*)d_out;

  float* ws     = (float*)d_ws;
  float* emis   = ws;                                   // ROWS * 4 floats (1 MB)
  float* chunkM = ws + (size_t)ROWS * EMIS_STRIDE;      // B * NCH * 9 floats (36 KB)

  // 4096 tiles of 16 rows, 8 waves (tiles) per 256-thread block -> 512 blocks
  crf_emis_wmma<<<512, 256, 0, stream>>>(x, W, bias, emis);
  // B*NCH = 1024 chunk threads
  crf_chunk_scan<<<(Bsz * NCH + 255) / 256, 256, 0, stream>>>(emis, mask, trans, chunkM);
  // one wave32 per batch
  crf_finalize<<<1, 512, 0, stream>>>(emis, y, mask, start_t, end_t, trans, chunkM, out);
}